// SCRE_56057913147946
// MI455X (gfx1250) — compile-verified
//
#include <hip/hip_runtime.h>

#define D_FEAT 128
#define NREL 3
#define WAVES_PER_BLOCK 8
#define ROWS_PER_GROUP 8   // edges gathered per TDM descriptor (32-bit gather indices -> max 8)

typedef __attribute__((ext_vector_type(4))) unsigned int v4u;
typedef __attribute__((ext_vector_type(8))) int          v8i;
typedef __attribute__((ext_vector_type(4))) int          v4i;

// ---------------------------------------------------------------- zero ws ---
__global__ void SCRE_zero_ws(float* __restrict__ ws, long n) {
    long i = (long)blockIdx.x * blockDim.x + threadIdx.x;
    long stride = (long)gridDim.x * blockDim.x;
    for (; i < n; i += stride) ws[i] = 0.0f;
}

// -------------------------------------------------- TDM gather descriptor ---
// Gather-mode D# (CDNA5 ISA): one instruction DMAs up to
// 8 rows (32-bit indices) of x[*, 0:128] from global into LDS, tracked by
// TENSORcnt.  'col' holds the row index in lanes 0..7.
__device__ __forceinline__ void issue_tdm_gather(const float* __restrict__ x,
                                                 int col, int nvalid, int N,
                                                 unsigned lds_byte_off) {
    unsigned long long xaddr = (unsigned long long)(const void*)x;

    v4u g0;
    // count=1 (valid), gather_index_size=1 (32-bit), gather_mode=1
    g0.x = 0xC0000001u;
    g0.y = lds_byte_off;                                   // lds_addr
    g0.z = (unsigned)xaddr;                                // global_addr[31:0]
    g0.w = ((unsigned)(xaddr >> 32) & 0x01FFFFFFu)         // global_addr[56:32]
           | 0x80000000u;                                  // type=2 ("image")

    v8i g1;
    g1[0] = 0x00020000;                                    // data_size=2 (4B), mask=0
    g1[1] = (int)(((unsigned)D_FEAT & 0xFFFFu) << 16);     // tensor_dim0 lo16 @[63:48]
    g1[2] = (int)(((unsigned)D_FEAT >> 16) |               // tensor_dim0 hi16 @[79:64]
                  (((unsigned)N & 0xFFFFu) << 16));        // tensor_dim1 lo16 @[95:80]
    g1[3] = (int)((((unsigned)N >> 16) & 0xFFFFu) |        // tensor_dim1 hi16 @[111:96]
                  (((unsigned)D_FEAT) << 16));             // tile_dim0 @[127:112]
    g1[4] = nvalid & 0xFFFF;                               // tile_dim1 = #valid indices
    g1[5] = D_FEAT;                                        // tensor_dim0_stride lo32
    g1[6] = 0;                                             // stride hi / dim1_stride (ignored)
    g1[7] = 0;

    v4i g2 = { __builtin_amdgcn_readlane(col, 0),
               __builtin_amdgcn_readlane(col, 1),
               __builtin_amdgcn_readlane(col, 2),
               __builtin_amdgcn_readlane(col, 3) };
    v4i g3 = { __builtin_amdgcn_readlane(col, 4),
               __builtin_amdgcn_readlane(col, 5),
               __builtin_amdgcn_readlane(col, 6),
               __builtin_amdgcn_readlane(col, 7) };

    v8i g4 = { 0, 0, 0, 0, 0, 0, 0, 0 };                   // beyond-D# extra group (clang-23
                                                           // 6-arg form); zero-filled

    __builtin_amdgcn_tensor_load_to_lds(g0, g1, g2, g3, g4, 0);
}

// ------------------------------------------------------------- scatter-add --
__global__ __launch_bounds__(256)
void SCRE_scatter_tdm(const float* __restrict__ x,
                      const int*   __restrict__ ei,   // [2,E]: ei[e]=row, ei[E+e]=col
                      const int*   __restrict__ et,   // [E]
                      float* __restrict__ sums,       // [NREL,N,D]
                      float* __restrict__ cnts,       // [NREL,N]
                      int N, int E) {
    // per-wave double-buffered LDS staging: 8 rows x 128 f32 x 2 bufs
    __shared__ float sbuf[WAVES_PER_BLOCK][2][ROWS_PER_GROUP * D_FEAT];

    const int lane   = threadIdx.x & 31;
    const int wid    = threadIdx.x >> 5;
    const int gwave  = blockIdx.x * WAVES_PER_BLOCK + wid;
    const int nwaves = gridDim.x * WAVES_PER_BLOCK;
    const int G      = (E + ROWS_PER_GROUP - 1) / ROWS_PER_GROUP;

    int g = gwave;
    if (g >= G) return;

    // --- prologue: metadata + TDM issue for the first group -----------------
    int cur_row = 0, cur_type = 0, col = 0;
    {
        int e = g * ROWS_PER_GROUP + lane;
        if (lane < ROWS_PER_GROUP && e < E) {
            cur_row  = ei[e];
            cur_type = et[e];
            col      = ei[(long)E + e];
        }
    }
    int buf = 0;
    {
        int nv = E - g * ROWS_PER_GROUP; if (nv > ROWS_PER_GROUP) nv = ROWS_PER_GROUP;
        unsigned lds0 = (unsigned)(unsigned long long)(const void*)&sbuf[wid][buf][0];
        issue_tdm_gather(x, col, nv, N, lds0);
    }

    while (g < G) {
        const int gn = g + nwaves;
        int nxt_row = 0, nxt_type = 0;

        if (gn < G) {   // issue next group's gather into the other buffer
            int ncol = 0;
            int en = gn * ROWS_PER_GROUP + lane;
            if (lane < ROWS_PER_GROUP && en < E) {
                nxt_row  = ei[en];
                nxt_type = et[en];
                ncol     = ei[(long)E + en];
            }
            int nv = E - gn * ROWS_PER_GROUP; if (nv > ROWS_PER_GROUP) nv = ROWS_PER_GROUP;
            unsigned lds1 = (unsigned)(unsigned long long)(const void*)&sbuf[wid][buf ^ 1][0];
            issue_tdm_gather(x, ncol, nv, N, lds1);
            __builtin_amdgcn_s_wait_tensorcnt(1);   // current group's DMA done
        } else {
            __builtin_amdgcn_s_wait_tensorcnt(0);
        }

        // --- process current group from LDS --------------------------------
        int nv = E - g * ROWS_PER_GROUP; if (nv > ROWS_PER_GROUP) nv = ROWS_PER_GROUP;
        const float4* rowbuf = (const float4*)&sbuf[wid][buf][0];
#pragma unroll
        for (int j = 0; j < ROWS_PER_GROUP; ++j) {
            if (j < nv) {
                int r = __shfl(cur_row,  j, 32);
                int t = __shfl(cur_type, j, 32);
                float4 v = rowbuf[j * (D_FEAT / 4) + lane];   // ds_load_b128
                float* dst = sums + (((long)t * N + r) * D_FEAT) + lane * 4;
                atomicAdd(dst + 0, v.x);
                atomicAdd(dst + 1, v.y);
                atomicAdd(dst + 2, v.z);
                atomicAdd(dst + 3, v.w);
                if (lane == 0) atomicAdd(cnts + (long)t * N + r, 1.0f);
            }
        }

        cur_row = nxt_row; cur_type = nxt_type;
        g = gn; buf ^= 1;
    }
}

// --------------------------------------------------------------- finalize ---
__global__ __launch_bounds__(256)
void SCRE_finalize(const float* __restrict__ x,
                   const float* __restrict__ sums,
                   const float* __restrict__ cnts,
                   float* __restrict__ out, int N) {
    long i = (long)blockIdx.x * blockDim.x + threadIdx.x;
    long total = (long)N * D_FEAT;
    if (i >= total) return;
    int node = (int)(i >> 7);          // D_FEAT = 128
    int feat = (int)(i & 127);

    float ctx = 0.0f, rc = 0.0f;
#pragma unroll
    for (int r = 0; r < NREL; ++r) {
        float c = cnts[(long)r * N + node];                      // wave-uniform -> s_load
        float s = sums[((long)r * N + node) * D_FEAT + feat];
        ctx += s / fmaxf(c, 1.0f);
        rc  += (c > 0.0f) ? 1.0f : 0.0f;
    }
    out[i] = x[i] - ctx / fmaxf(rc, 1.0f);
}

// ------------------------------------------------------------------ launch --
extern "C" void kernel_launch(void* const* d_in, const int* in_sizes, int n_in,
                              void* d_out, int out_size, void* d_ws, size_t ws_size,
                              hipStream_t stream) {
    const float* x  = (const float*)d_in[0];
    const int*   ei = (const int*)d_in[1];
    const int*   et = (const int*)d_in[2];
    float* out = (float*)d_out;

    const int ND = in_sizes[0];
    const int N  = ND / D_FEAT;
    const int E  = in_sizes[2];

    float* sums = (float*)d_ws;                               // [NREL,N,D]
    float* cnts = sums + (long)NREL * N * D_FEAT;             // [NREL,N]
    const long nz = (long)NREL * N * D_FEAT + (long)NREL * N;

    SCRE_zero_ws<<<2048, 256, 0, stream>>>((float*)d_ws, nz);

    // 1024 blocks * 8 waves -> each wave pipelines ~9 TDM gather groups
    SCRE_scatter_tdm<<<1024, 256, 0, stream>>>(x, ei, et, sums, cnts, N, E);

    const int total = ND;
    SCRE_finalize<<<(total + 255) / 256, 256, 0, stream>>>(x, sums, cnts, out, N);
}